// mLSTM_69965017252758
// MI455X (gfx1250) — compile-verified
//
#include <hip/hip_runtime.h>
#include <math.h>

typedef __attribute__((ext_vector_type(2))) float v2f;
typedef __attribute__((ext_vector_type(4))) float v4f;
typedef __attribute__((ext_vector_type(8))) float v8f;

#define T_DIM 512
#define B_DIM 8
#define D_DIM 512
#define M_DIM 128
#define NROW  (T_DIM * B_DIM)          // 4096 rows of X
#define KSCALE 0.08838834764831843f    // 1/sqrt(128)

// ---------------------------------------------------------------------------
// Kernel 1: projections  Y_w = act(X @ W_w + b_w)  for w in {q,k,v,o}
// X: (4096, 512) f32 row-major; W: (512, 128) f32 row-major.
// Exact f32 math via V_WMMA_F32_16X16X4_F32 (cost is negligible vs HBM stream,
// so we keep full precision). One block = one 16-row tile; 8 waves, each wave
// owns 4 of the 32 (weight, 16-col) tiles. K loop: 512 / 4 = 128 steps.
// Weights (1 MB total) stay L2-resident across all 256 blocks.
// ---------------------------------------------------------------------------
__global__ __launch_bounds__(256) void mlstm_proj_kernel(
    const float* __restrict__ x,
    const float* __restrict__ Wq, const float* __restrict__ bq,
    const float* __restrict__ Wk, const float* __restrict__ bk,
    const float* __restrict__ Wv, const float* __restrict__ bv,
    const float* __restrict__ Wo, const float* __restrict__ bo,
    float* __restrict__ wsq, float* __restrict__ wsk,
    float* __restrict__ wsv, float* __restrict__ wso)
{
    const int lane = threadIdx.x & 31;
    const int wave = threadIdx.x >> 5;
    const int half = lane >> 4;    // 0: lanes 0-15, 1: lanes 16-31
    const int mlo  = lane & 15;

    const int row0 = blockIdx.x * 16;
    const int widx = wave >> 1;            // 0=q 1=k 2=v 3=o (wave-uniform)
    const int colBase = (wave & 1) * 64;   // this wave covers cols colBase..colBase+63

    const float* W;  const float* bias;  float* dst;
    switch (widx) {
        case 0: W = Wq; bias = bq; dst = wsq; break;
        case 1: W = Wk; bias = bk; dst = wsk; break;
        case 2: W = Wv; bias = bv; dst = wsv; break;
        default:W = Wo; bias = bo; dst = wso; break;
    }

    // A-fragment source: row (row0+mlo), K pair starting at 2*half.
    const float* xrow = x + (size_t)(row0 + mlo) * D_DIM + 2 * half;

    v8f acc0 = {}, acc1 = {}, acc2 = {}, acc3 = {};

    for (int k0 = 0; k0 < D_DIM; k0 += 4) {
        // A 16x4 f32 fragment: lane m<16 holds (M=m, K=k0+0/1); lanes 16-31 K=k0+2/3
        v2f a = *(const v2f*)(xrow + k0);

        // B 4x16 fragments (mirrored layout: vgpr0 = K rows {0,2}, vgpr1 = {1,3})
        const float* Wr0 = W + (size_t)(k0 + 2 * half) * M_DIM + colBase + mlo;
        const float* Wr1 = Wr0 + M_DIM;

        v2f b0; b0[0] = Wr0[ 0]; b0[1] = Wr1[ 0];
        v2f b1; b1[0] = Wr0[16]; b1[1] = Wr1[16];
        v2f b2; b2[0] = Wr0[32]; b2[1] = Wr1[32];
        v2f b3; b3[0] = Wr0[48]; b3[1] = Wr1[48];

        acc0 = __builtin_amdgcn_wmma_f32_16x16x4_f32(false, a, false, b0, (short)0, acc0, false, false);
        acc1 = __builtin_amdgcn_wmma_f32_16x16x4_f32(false, a, false, b1, (short)0, acc1, false, false);
        acc2 = __builtin_amdgcn_wmma_f32_16x16x4_f32(false, a, false, b2, (short)0, acc2, false, false);
        acc3 = __builtin_amdgcn_wmma_f32_16x16x4_f32(false, a, false, b3, (short)0, acc3, false, false);
    }

    // Epilogue: C/D layout -> element (M = r + 8*half, N = mlo) in vgpr r.
    v8f accs[4] = {acc0, acc1, acc2, acc3};
#pragma unroll
    for (int t = 0; t < 4; ++t) {
        const int col = colBase + t * 16 + mlo;
        const float bv_ = bias[col];
#pragma unroll
        for (int r = 0; r < 8; ++r) {
            const int m = r + 8 * half;
            float y = accs[t][r] + bv_;
            if (widx == 1)      y *= KSCALE;                 // k: (xWk+bk)/sqrt(M)
            else if (widx == 3) y = 1.0f / (1.0f + expf(-y)); // o: sigmoid
            dst[(size_t)(row0 + m) * M_DIM + col] = y;
        }
    }
}

// ---------------------------------------------------------------------------
// Kernel 2: per-(t,b) gated state update + normalized readout (HBM-bound).
// One 256-thread block per (t,b). Streams 64KB in + 64KB out as 128-bit
// NON-TEMPORAL accesses (512 MB stream, zero reuse -> don't thrash 192 MB L2),
// fusing n update, n.q, C = f*C_prev + i*v(x)k, htacc = C.q, h = sig(o)*htacc/den.
// ---------------------------------------------------------------------------
__global__ __launch_bounds__(256) void mlstm_update_kernel(
    const float* __restrict__ x,
    const float* __restrict__ C_prev,
    const float* __restrict__ n_prev,
    const float* __restrict__ wi, const float* __restrict__ bi,
    const float* __restrict__ wf, const float* __restrict__ bf,
    const float* __restrict__ wsq, const float* __restrict__ wsk,
    const float* __restrict__ wsv, const float* __restrict__ wso,
    float* __restrict__ out_ht, float* __restrict__ out_C, float* __restrict__ out_n)
{
    __shared__ float s_q[M_DIM], s_k[M_DIM], s_v[M_DIM], s_ht[M_DIM];
    __shared__ float red0[256], red1[256];
    __shared__ float s_it, s_ft, s_invden;

    const int tid  = threadIdx.x;
    const int lane = tid & 31;
    const int tb   = blockIdx.x;               // (t*B + b), matches all layouts

    // ---- gates: i = exp(x.wi + bi), f = sigmoid(x.wf + bf) -----------------
    const float* xrow = x + (size_t)tb * D_DIM;
    float pi = xrow[tid] * wi[tid] + xrow[tid + 256] * wi[tid + 256];
    float pf = xrow[tid] * wf[tid] + xrow[tid + 256] * wf[tid + 256];
    red0[tid] = pi; red1[tid] = pf;
    __syncthreads();
    for (int s = 128; s > 0; s >>= 1) {
        if (tid < s) { red0[tid] += red0[tid + s]; red1[tid] += red1[tid + s]; }
        __syncthreads();
    }
    if (tid == 0) {
        s_it = expf(red0[0] + bi[0]);
        s_ft = 1.0f / (1.0f + expf(-(red1[0] + bf[0])));
    }
    __syncthreads();
    const float it = s_it, ft = s_ft;

    // ---- n = f*n_prev + i*k ; den = max(|n.q|, 1) --------------------------
    float pd = 0.0f;
    if (tid < M_DIM) {
        const size_t v0 = (size_t)tb * M_DIM + tid;
        const float qj = wsq[v0];
        const float kj = wsk[v0];
        const float nj = ft * n_prev[v0] + it * kj;
        out_n[v0] = nj;
        s_q[tid] = qj; s_k[tid] = kj; s_v[tid] = wsv[v0];
        pd = nj * qj;
    }
    red0[tid] = pd;
    __syncthreads();
    for (int s = 128; s > 0; s >>= 1) {
        if (tid < s) red0[tid] += red0[tid + s];
        __syncthreads();
    }
    if (tid == 0) s_invden = 1.0f / fmaxf(fabsf(red0[0]), 1.0f);
    __syncthreads();

    // ---- stream C: 16384 floats = 4096 x 128-bit; 16 vectors per thread ----
    const v4f* Cp4 = (const v4f*)(C_prev + (size_t)tb * M_DIM * M_DIM);
    v4f*       Co4 = (v4f*)(out_C + (size_t)tb * M_DIM * M_DIM);

    const v4f qv = *(const v4f*)&s_q[lane * 4];
    const v4f kv = *(const v4f*)&s_k[lane * 4];

#pragma unroll
    for (int s = 0; s < 16; ++s) {
        const int c   = tid + 256 * s;        // 128-bit index within tile
        const int row = c >> 5;               // wave-uniform: row = wave + 8*s
        const float iv = it * s_v[row];

        v4f cp = __builtin_nontemporal_load(&Cp4[c]);   // TH=NT: no-reuse stream
        v4f o  = ft * cp + iv * kv;
        __builtin_nontemporal_store(o, &Co4[c]);        // TH=NT: write-once stream

        v4f pq = o * qv;
        float p = pq.x + pq.y + pq.z + pq.w;
#pragma unroll
        for (int off = 16; off > 0; off >>= 1)
            p += __shfl_xor(p, off, 32);
        if (lane == 0) s_ht[row] = p;         // each (wave,s) owns a unique row
    }
    __syncthreads();

    // ---- h = sigmoid(o) * (C.q) / den --------------------------------------
    if (tid < M_DIM) {
        const size_t v0 = (size_t)tb * M_DIM + tid;
        out_ht[v0] = wso[v0] * s_ht[tid] * s_invden;
    }
}

// ---------------------------------------------------------------------------
extern "C" void kernel_launch(void* const* d_in, const int* in_sizes, int n_in,
                              void* d_out, int out_size, void* d_ws, size_t ws_size,
                              hipStream_t stream)
{
    const float* x      = (const float*)d_in[0];
    const float* C_prev = (const float*)d_in[1];
    const float* n_prev = (const float*)d_in[2];
    const float* Wq = (const float*)d_in[3];  const float* bq = (const float*)d_in[4];
    const float* Wk = (const float*)d_in[5];  const float* bk = (const float*)d_in[6];
    const float* Wv = (const float*)d_in[7];  const float* bv = (const float*)d_in[8];
    const float* wi = (const float*)d_in[9];  const float* bi = (const float*)d_in[10];
    const float* wf = (const float*)d_in[11]; const float* bf = (const float*)d_in[12];
    const float* Wo = (const float*)d_in[13]; const float* bo = (const float*)d_in[14];

    float* ws  = (float*)d_ws;
    float* wsq = ws;
    float* wsk = wsq + (size_t)NROW * M_DIM;
    float* wsv = wsk + (size_t)NROW * M_DIM;
    float* wso = wsv + (size_t)NROW * M_DIM;   // total 8 MB of d_ws

    float* out_ht = (float*)d_out;                               // (T,B,M)
    float* out_C  = out_ht + (size_t)NROW * M_DIM;               // (T,B,M,M)
    float* out_n  = out_C  + (size_t)NROW * M_DIM * M_DIM;       // (T,B,M)

    mlstm_proj_kernel<<<dim3(NROW / 16), dim3(256), 0, stream>>>(
        x, Wq, bq, Wk, bk, Wv, bv, Wo, bo, wsq, wsk, wsv, wso);

    mlstm_update_kernel<<<dim3(NROW), dim3(256), 0, stream>>>(
        x, C_prev, n_prev, wi, bi, wf, bf, wsq, wsk, wsv, wso,
        out_ht, out_C, out_n);
}